// GAT_Graph_35158602285146
// MI455X (gfx1250) — compile-verified
//
#include <hip/hip_runtime.h>
#include <math.h>

#define N_NODES   50000
#define N_EDGES   500000
#define E_TOT     (N_EDGES + N_NODES)   // self-loops appended
#define IN_CH     128
#define HID       64
#define HEADS     4
#define FDIM      256                   // HEADS*HID
#define N_GRAPHS  64
#define OUT_CH    16
#define NEG_SLOPE 0.2f

typedef __attribute__((ext_vector_type(2))) float v2f;
typedef __attribute__((ext_vector_type(8))) float v8f;

// ---------------------------------------------------------------------------
// GEMM: C[M x 256] = A[M x K] @ W[K x 256], fp32 WMMA 16x16x4.
// Block = 256 threads = 8 waves; wave w handles column tile (blockIdx.y*8 + w).
// grid = (M/16, 2). M = 50000 is an exact multiple of 16.
// ---------------------------------------------------------------------------
__global__ void gemm_wmma_f32(const float* __restrict__ A,
                              const float* __restrict__ W,
                              float* __restrict__ C, int K) {
    const int wave  = threadIdx.x >> 5;
    const int lane  = threadIdx.x & 31;
    const int tileM = blockIdx.x;
    const int tileN = blockIdx.y * 8 + wave;          // 0..15
    const int row   = tileM * 16 + (lane & 15);
    const int col   = tileN * 16 + (lane & 15);
    const int khalf = (lane >> 4) * 2;                // lanes 16..31 take K+2,K+3

    const float* __restrict__ arow = A + (size_t)row * K;
    v8f acc = {};
    for (int k0 = 0; k0 < K; k0 += 4) {
        const int kb = k0 + khalf;
        v2f a, b;
        a.x = arow[kb];
        a.y = arow[kb + 1];
        b.x = W[(size_t)kb       * FDIM + col];
        b.y = W[(size_t)(kb + 1) * FDIM + col];
        acc = __builtin_amdgcn_wmma_f32_16x16x4_f32(
                  false, a, false, b, (short)0, acc, false, false);
    }
    // D layout: VGPR v -> row (tileM*16 + v + (lane>=16 ? 8 : 0)), col = lane&15
    const int rbase = tileM * 16 + ((lane >> 4) << 3);
#pragma unroll
    for (int v = 0; v < 8; ++v)
        C[(size_t)(rbase + v) * FDIM + col] = acc[v];
}

// ---------------------------------------------------------------------------
// Attention scores: one wave per node, lane l covers channels [8l, 8l+8)
// (head = l/8). Reduce the 8 lanes of each head with shuffles.
// ---------------------------------------------------------------------------
__global__ void scores_kernel(const float* __restrict__ h,
                              const float* __restrict__ asrc,
                              const float* __restrict__ adst,
                              float* __restrict__ s_src,
                              float* __restrict__ s_dst) {
    const int node = blockIdx.x * (blockDim.x >> 5) + (threadIdx.x >> 5);
    if (node >= N_NODES) return;
    const int lane = threadIdx.x & 31;
    const float* __restrict__ hp = h    + (size_t)node * FDIM + lane * 8;
    const float* __restrict__ ap = asrc + lane * 8;   // (H,HID) flat == channel order
    const float* __restrict__ bp = adst + lane * 8;
    float ps = 0.f, pd = 0.f;
#pragma unroll
    for (int i = 0; i < 8; ++i) {
        const float v = hp[i];
        ps += v * ap[i];
        pd += v * bp[i];
    }
#pragma unroll
    for (int off = 1; off < 8; off <<= 1) {
        ps += __shfl_xor(ps, off, 32);
        pd += __shfl_xor(pd, off, 32);
    }
    if ((lane & 7) == 0) {
        const int head = lane >> 3;
        s_src[node * HEADS + head] = ps;
        s_dst[node * HEADS + head] = pd;
    }
}

__global__ void fill_kernel(float* __restrict__ p, float val, size_t n) {
    const size_t i = (size_t)blockIdx.x * blockDim.x + threadIdx.x;
    if (i < n) p[i] = val;
}

__device__ __forceinline__ void edge_endpoints(const int* __restrict__ ei,
                                               int e, int& src, int& dst) {
    if (e < N_EDGES) { src = ei[e]; dst = ei[N_EDGES + e]; }
    else             { src = dst = e - N_EDGES; }          // self-loop
}

// Pass A: e = leaky_relu(s_src[src]+s_dst[dst]); running max into m[dst]
__global__ void edge_max_kernel(const int* __restrict__ ei,
                                const float* __restrict__ s_src,
                                const float* __restrict__ s_dst,
                                float* __restrict__ e_buf,
                                float* __restrict__ m) {
    const int e = blockIdx.x * blockDim.x + threadIdx.x;
    if (e >= E_TOT) return;
    int src, dst; edge_endpoints(ei, e, src, dst);
#pragma unroll
    for (int hh = 0; hh < HEADS; ++hh) {
        const float x = s_src[src * HEADS + hh] + s_dst[dst * HEADS + hh];
        const float v = x > 0.f ? x : NEG_SLOPE * x;
        e_buf[(size_t)e * HEADS + hh] = v;
        atomicMax(&m[dst * HEADS + hh], v);   // -> global_atomic_max_num_f32
    }
}

// Pass B: ex = exp(e - m[dst]); denom[dst] += ex
__global__ void edge_exp_kernel(const int* __restrict__ ei,
                                const float* __restrict__ m,
                                float* __restrict__ e_buf,
                                float* __restrict__ denom) {
    const int e = blockIdx.x * blockDim.x + threadIdx.x;
    if (e >= E_TOT) return;
    int src, dst; edge_endpoints(ei, e, src, dst);
    (void)src;
#pragma unroll
    for (int hh = 0; hh < HEADS; ++hh) {
        const float ex = __expf(e_buf[(size_t)e * HEADS + hh] - m[dst * HEADS + hh]);
        e_buf[(size_t)e * HEADS + hh] = ex;
        atomicAdd(&denom[dst * HEADS + hh], ex);
    }
}

// Pass C: out[dst] += alpha * h[src]; one wave per edge, 8 ch/lane (coalesced)
__global__ void edge_aggr_kernel(const int* __restrict__ ei,
                                 const float* __restrict__ e_buf,
                                 const float* __restrict__ denom,
                                 const float* __restrict__ h,
                                 float* __restrict__ out) {
    const int e = blockIdx.x * (blockDim.x >> 5) + (threadIdx.x >> 5);
    if (e >= E_TOT) return;
    const int lane = threadIdx.x & 31;
    int src, dst; edge_endpoints(ei, e, src, dst);
    const int head = lane >> 3;
    const float alpha = e_buf[(size_t)e * HEADS + head] /
                        (denom[dst * HEADS + head] + 1e-16f);
    const float* __restrict__ hp = h   + (size_t)src * FDIM + lane * 8;
    float*       __restrict__ op = out + (size_t)dst * FDIM + lane * 8;
#pragma unroll
    for (int i = 0; i < 8; ++i)
        atomicAdd(&op[i], alpha * hp[i]);                 // global_atomic_add_f32
}

// out = relu(out + bias), in place
__global__ void bias_relu_kernel(float* __restrict__ x, const float* __restrict__ b) {
    const size_t i = (size_t)blockIdx.x * blockDim.x + threadIdx.x;
    if (i >= (size_t)N_NODES * FDIM) return;
    const float v = x[i] + b[i & (FDIM - 1)];
    x[i] = v > 0.f ? v : 0.f;
}

// global_add_pool: batch is sorted -> accumulate runs, flush on graph change
#define POOL_CHUNK 128
__global__ void pool_kernel(const float* __restrict__ h,
                            const int* __restrict__ batch,
                            float* __restrict__ pooled) {
    const int c  = threadIdx.x;                 // 256 threads == 256 channels
    const int n0 = blockIdx.x * POOL_CHUNK;
    int n1 = n0 + POOL_CHUNK; if (n1 > N_NODES) n1 = N_NODES;
    float acc = 0.f;
    int cur = batch[n0];
    for (int n = n0; n < n1; ++n) {
        const int g = batch[n];
        if (g != cur) {
            atomicAdd(&pooled[cur * FDIM + c], acc);
            acc = 0.f; cur = g;
        }
        acc += h[(size_t)n * FDIM + c];
    }
    atomicAdd(&pooled[cur * FDIM + c], acc);
}

// out[64x16] = pooled[64x256] @ Wc[256x16] + bc   (4 waves, one 16x16 tile each)
__global__ void final_gemm_kernel(const float* __restrict__ P,
                                  const float* __restrict__ Wc,
                                  const float* __restrict__ bc,
                                  float* __restrict__ out) {
    const int wave  = threadIdx.x >> 5;          // 0..3 -> M tiles
    const int lane  = threadIdx.x & 31;
    const int row   = wave * 16 + (lane & 15);
    const int col   = lane & 15;
    const int khalf = (lane >> 4) * 2;
    v8f acc = {};
    for (int k0 = 0; k0 < FDIM; k0 += 4) {
        const int kb = k0 + khalf;
        v2f a, b;
        a.x = P[row * FDIM + kb];
        a.y = P[row * FDIM + kb + 1];
        b.x = Wc[kb * OUT_CH + col];
        b.y = Wc[(kb + 1) * OUT_CH + col];
        acc = __builtin_amdgcn_wmma_f32_16x16x4_f32(
                  false, a, false, b, (short)0, acc, false, false);
    }
    const int rbase = wave * 16 + ((lane >> 4) << 3);
    const float bias = bc[col];
#pragma unroll
    for (int v = 0; v < 8; ++v)
        out[(rbase + v) * OUT_CH + col] = acc[v] + bias;
}

// ---------------------------------------------------------------------------
extern "C" void kernel_launch(void* const* d_in, const int* in_sizes, int n_in,
                              void* d_out, int out_size, void* d_ws, size_t ws_size,
                              hipStream_t stream) {
    (void)in_sizes; (void)n_in; (void)out_size; (void)ws_size;
    const float* x     = (const float*)d_in[0];
    // d_in[1] = edge_attr (unused by reference)
    const int*   ei    = (const int*)d_in[2];     // [2, E]
    const int*   batch = (const int*)d_in[3];
    const float* W[3]    = {(const float*)d_in[4],  (const float*)d_in[8],  (const float*)d_in[12]};
    const float* asrc[3] = {(const float*)d_in[5],  (const float*)d_in[9],  (const float*)d_in[13]};
    const float* adst[3] = {(const float*)d_in[6],  (const float*)d_in[10], (const float*)d_in[14]};
    const float* bias[3] = {(const float*)d_in[7],  (const float*)d_in[11], (const float*)d_in[15]};
    const float* Wc = (const float*)d_in[16];
    const float* bc = (const float*)d_in[17];
    float* out = (float*)d_out;

    // workspace layout
    char* ws = (char*)d_ws;
    size_t off = 0;
    auto alloc = [&](size_t bytes) { float* p = (float*)(ws + off); off += (bytes + 255) & ~(size_t)255; return p; };
    float* buf_h   = alloc((size_t)N_NODES * FDIM * 4);  // GEMM output h
    float* buf_agg = alloc((size_t)N_NODES * FDIM * 4);  // aggregated layer output
    float* s_src   = alloc((size_t)N_NODES * HEADS * 4);
    float* s_dst   = alloc((size_t)N_NODES * HEADS * 4);
    float* m_buf   = alloc((size_t)N_NODES * HEADS * 4);
    float* den_buf = alloc((size_t)N_NODES * HEADS * 4);
    float* e_buf   = alloc((size_t)E_TOT   * HEADS * 4);
    float* pooled  = alloc((size_t)N_GRAPHS * FDIM * 4);

    const size_t nF  = (size_t)N_NODES * FDIM;        // 12.8M
    const size_t nNH = (size_t)N_NODES * HEADS;       // 200K

    const float* cur_in = x;
    int K = IN_CH;
    for (int layer = 0; layer < 3; ++layer) {
        // h = cur_in @ W  (fp32 WMMA)
        gemm_wmma_f32<<<dim3(N_NODES / 16, 2), 256, 0, stream>>>(cur_in, W[layer], buf_h, K);
        // attention scores
        scores_kernel<<<(N_NODES + 7) / 8, 256, 0, stream>>>(buf_h, asrc[layer], adst[layer], s_src, s_dst);
        // init accumulators
        fill_kernel<<<(unsigned)((nF  + 255) / 256), 256, 0, stream>>>(buf_agg, 0.f, nF);
        fill_kernel<<<(unsigned)((nNH + 255) / 256), 256, 0, stream>>>(m_buf, -INFINITY, nNH);
        fill_kernel<<<(unsigned)((nNH + 255) / 256), 256, 0, stream>>>(den_buf, 0.f, nNH);
        // segment softmax + weighted aggregation
        edge_max_kernel <<<(E_TOT + 255) / 256, 256, 0, stream>>>(ei, s_src, s_dst, e_buf, m_buf);
        edge_exp_kernel <<<(E_TOT + 255) / 256, 256, 0, stream>>>(ei, m_buf, e_buf, den_buf);
        edge_aggr_kernel<<<(E_TOT + 7) / 8,     256, 0, stream>>>(ei, e_buf, den_buf, buf_h, buf_agg);
        // bias + relu
        bias_relu_kernel<<<(unsigned)((nF + 255) / 256), 256, 0, stream>>>(buf_agg, bias[layer]);
        cur_in = buf_agg;
        K = FDIM;
    }

    // global_add_pool + classifier head
    fill_kernel<<<(N_GRAPHS * FDIM + 255) / 256, 256, 0, stream>>>(pooled, 0.f, (size_t)N_GRAPHS * FDIM);
    pool_kernel<<<(N_NODES + POOL_CHUNK - 1) / POOL_CHUNK, 256, 0, stream>>>(buf_agg, batch, pooled);
    final_gemm_kernel<<<1, 128, 0, stream>>>(pooled, Wc, bc, out);
}